// lstm_reg_53618371723875
// MI455X (gfx1250) — compile-verified
//
#include <hip/hip_runtime.h>

// Problem sizes (match reference)
#define S_LEN 2048
#define B_SZ  1024
#define H_SZ  4
#define NL    6

typedef float v2f __attribute__((ext_vector_type(2)));
typedef float v8f __attribute__((ext_vector_type(8)));

// Single-instruction tanh on CDNA5 (v_tanh_f32); cheap fallback otherwise.
__device__ __forceinline__ float fast_tanh(float x) {
#if __has_builtin(__builtin_amdgcn_tanhf)
    return __builtin_amdgcn_tanhf(x);
#else
    return 2.0f * __builtin_amdgcn_rcpf(1.0f + __expf(-2.0f * x)) - 1.0f;
#endif
}

// One wave (32 lanes) owns a 16-wide batch tile through the 6-layer stack and
// all 2048 timesteps.  Per LSTM cell (two chained f32 16x16x4 WMMAs):
//   D1 = WMMA( A=W_ih[l] (16x4, rows permuted+prescaled), B=input (4x16), C=bias )
//   G  = WMMA( A=W_hh[l] (same row treatment),            B=h_prev (4x16), C=D1 )
//
// A-row permutation perm(m) = ((m&7)>>1)*4 + (m&1) + (m>=8 ? 2 : 0) puts
//   rows 0..7  = i0,i1,f0,f1,g0,g1,o0,o1   (-> low  lanes, hidden units 0,1)
//   rows 8..15 = i2,i3,f2,f3,g2,g3,o2,o3   (-> high lanes, hidden units 2,3)
// so every lane holds a complete i/f/g/o set for its two hidden units of its
// batch column: the cell update is lane-local and (h0,h1 | h2,h3) lands
// directly in the B-matrix layout (VGPR0 = K0|K2, VGPR1 = K1|K3) for the next
// WMMA.  Sigmoid rows (i,f,o) of W and bias are prescaled by 0.5 so that
//   sigmoid(x) = 0.5*tanh(x/2) + 0.5  ==  fma(0.5, tanh(d), 0.5).
__global__ __launch_bounds__(32) void lstm_stack_wmma_kernel(
    const float* __restrict__ x,      // [S, B, 4]
    const float* __restrict__ w_ih,   // [NL, 16, 4]
    const float* __restrict__ w_hh,   // [NL, 16, 4]
    const float* __restrict__ b_ih,   // [NL, 16]
    const float* __restrict__ b_hh,   // [NL, 16]
    const float* __restrict__ reg_w,  // [1, 4]
    const float* __restrict__ reg_b,  // [1]
    float* __restrict__ out)          // [S, B, 1]
{
    const int  lane  = threadIdx.x & 31;
    const bool lo    = lane < 16;
    const int  m     = lane & 15;        // A row slot / batch column
    const int  k0    = lo ? 0 : 2;       // K pair handled by this half-wave
    const int  bbase = blockIdx.x * 16;  // batch tile base

    // Logical gate row this lane's A slot holds, and its prescale.
    const int   arow   = (((m & 7) >> 1) << 2) + (m & 1) + ((m >= 8) ? 2 : 0);
    const float ascale = (((m & 7) == 4) || ((m & 7) == 5)) ? 1.0f : 0.5f; // g rows unscaled

    v2f Aih[NL], Ahh[NL], Hb[NL];
    v8f Cb[NL];
    float cst[NL][2];   // two cell-state values per lane per layer

#pragma unroll
    for (int l = 0; l < NL; ++l) {
        const float* wi = w_ih + l * 64 + arow * 4 + k0;
        const float* wh = w_hh + l * 64 + arow * 4 + k0;
        Aih[l] = (v2f){ascale * wi[0], ascale * wi[1]};
        Ahh[l] = (v2f){ascale * wh[0], ascale * wh[1]};
#pragma unroll
        for (int r = 0; r < 8; ++r) {
            // D VGPR r holds logical gate row perm(r) (low) / perm(r+8) (high).
            const int   base   = ((r >> 1) << 2) + (r & 1);
            const int   gate   = base + (lo ? 0 : 2);
            const float bscale = (r == 4 || r == 5) ? 1.0f : 0.5f;
            Cb[l][r] = bscale * (b_ih[l * 16 + gate] + b_hh[l * 16 + gate]);
        }
        Hb[l]     = (v2f){0.f, 0.f};   // h0 = 0
        cst[l][0] = 0.f;
        cst[l][1] = 0.f;               // c0 = 0
    }

    // Head weights: lane-local pair of the 4-wide dot product.
    const float rwA = lo ? reg_w[0] : reg_w[2];
    const float rwB = lo ? reg_w[1] : reg_w[3];
    const float rb  = reg_b[0];

    // x in B layout: each lane loads its K-pair for its batch column.
    const float* xlane = x + ((size_t)(bbase + m) * H_SZ + k0);
    v2f xv = (v2f){xlane[0], xlane[1]};   // step 0, software-pipelined

    for (int s = 0; s < S_LEN; ++s) {
        v2f inB = xv;
        if (s + 1 < S_LEN) {              // prefetch next step's x behind compute
            const float* xp = xlane + (size_t)(s + 1) * B_SZ * H_SZ;
            xv = (v2f){xp[0], xp[1]};
        }

        float hA = 0.f, hB = 0.f;
#pragma unroll
        for (int l = 0; l < NL; ++l) {
            // pre = W_ih * input + bias ; gates = W_hh * h_prev + pre
            v8f gt = __builtin_amdgcn_wmma_f32_16x16x4_f32(
                false, Aih[l], false, inB, (short)0, Cb[l], false, false);
            gt = __builtin_amdgcn_wmma_f32_16x16x4_f32(
                false, Ahh[l], false, Hb[l], (short)0, gt, false, false);

            // act: r=0,1 -> i ; 2,3 -> f ; 4,5 -> g ; 6,7 -> o  (per lane).
            float act[8];
#pragma unroll
            for (int r = 0; r < 8; ++r) {
                float t = fast_tanh(gt[r]);
                act[r] = (r == 4 || r == 5) ? t : __builtin_fmaf(0.5f, t, 0.5f);
            }

            // Lane-local cell update for this lane's two hidden units.
            float cn0 = __builtin_fmaf(act[2], cst[l][0], act[0] * act[4]);
            float cn1 = __builtin_fmaf(act[3], cst[l][1], act[1] * act[5]);
            cst[l][0] = cn0;
            cst[l][1] = cn1;
            hA = act[6] * fast_tanh(cn0);
            hB = act[7] * fast_tanh(cn1);

            // (hA,hB) is already in B-matrix layout: VGPR0 = h0|h2, VGPR1 = h1|h3.
            v2f nb = (v2f){hA, hB};
            inB   = nb;     // feeds layer l+1 this timestep
            Hb[l] = nb;     // feeds layer l next timestep
        }

        // Linear head: pairwise partial per lane, one cross-half reduce.
        float part = __builtin_fmaf(rwA, hA, rwB * hB);
        float tot  = part + __shfl_xor(part, 16, 32);
        if (lo) out[(size_t)s * B_SZ + bbase + lane] = rb + tot;
    }
}

extern "C" void kernel_launch(void* const* d_in, const int* in_sizes, int n_in,
                              void* d_out, int out_size, void* d_ws, size_t ws_size,
                              hipStream_t stream) {
    const float* x     = (const float*)d_in[0];
    const float* w_ih  = (const float*)d_in[1];
    const float* w_hh  = (const float*)d_in[2];
    const float* b_ih  = (const float*)d_in[3];
    const float* b_hh  = (const float*)d_in[4];
    const float* reg_w = (const float*)d_in[5];
    const float* reg_b = (const float*)d_in[6];
    float* out = (float*)d_out;

    // 64 independent waves, each owning a 16-wide batch tile; one wave per block.
    lstm_stack_wmma_kernel<<<B_SZ / 16, 32, 0, stream>>>(
        x, w_ih, w_hh, b_ih, b_hh, reg_w, reg_b, out);
}